// GIPAConv2_4741643895399
// MI455X (gfx1250) — compile-verified
//
#include <hip/hip_runtime.h>

// ---------------- problem constants (match reference) ----------------
constexpr int N_NODES = 50000;     // divisible by 16 -> 3125 tiles
constexpr int N_EDGES = 1600000;   // divisible by 16 -> 100000 tiles
constexpr int F_IN    = 128;
constexpr int F_E     = 16;
constexpr int OUT     = 32;
constexpr float SLOPE = 0.2f;
constexpr float EPS   = 1e-9f;

typedef float v2f __attribute__((ext_vector_type(2)));
typedef float v8f __attribute__((ext_vector_type(8)));

// D = A(16x4) * B(4x16) + C, fp32, wave32.  A frag: lane L holds
// A[L&15][k0+2*(L>>4) + {0,1}]; B frag (mirror): B[k0+2*(L>>4)+{0,1}][L&15].
// C/D: vgpr v, lane L -> C[v + 8*(L>>4)][L&15].
#define WMMA4(acc, a, b)                                                     \
  acc = __builtin_amdgcn_wmma_f32_16x16x4_f32(false, (a), false, (b),        \
                                              (short)0, (acc), false, false)

__device__ __forceinline__ float leaky(float x) { return x >= 0.f ? x : SLOPE * x; }

__device__ __forceinline__ unsigned flipf(float f) {
  unsigned b = __float_as_uint(f);
  return (b & 0x80000000u) ? ~b : (b | 0x80000000u);
}
__device__ __forceinline__ float unflipf(unsigned u) {
  return __uint_as_float((u & 0x80000000u) ? (u ^ 0x80000000u) : ~u);
}

// ---------------- kernel 0: zero scratch (seg-max sentinel == 0u == -inf) ---
__global__ __launch_bounds__(256) void zero_kernel(float* p, long n) {
  long i = (long)blockIdx.x * 256 + threadIdx.x;
  if (i < n) p[i] = 0.f;
}

// ---------------- kernel 1: fused node GEMMs (one wave = 16 rows) -----------
// fc    = leaky(feat @ Wp) @ Wp2
// esrc  = feat @ Was ; edst = feat @ Wad ; dterm = feat @ Wapd + bapd
__global__ __launch_bounds__(32) void node_gemm_kernel(
    const float* __restrict__ feat,
    const float* __restrict__ Wp,  const float* __restrict__ Wp2,
    const float* __restrict__ Was, const float* __restrict__ Wad,
    const float* __restrict__ Wapd, const float* __restrict__ bapd,
    float* __restrict__ fc, float* __restrict__ esrc,
    float* __restrict__ edst, float* __restrict__ dterm) {
  const int lane = threadIdx.x, lo = lane & 15, hi = lane >> 4;
  const int base = blockIdx.x * 16;

  v8f accP[2] = {}, accAS[2] = {}, accAD[2] = {}, accAP[2] = {};
  for (int k0 = 0; k0 < F_IN; k0 += 4) {
    const int ka = k0 + 2 * hi;
    v2f a = *(const v2f*)(feat + (long)(base + lo) * F_IN + ka);
#pragma unroll
    for (int t = 0; t < 2; ++t) {
      const int n = lo + 16 * t;
      v2f b;
      b = (v2f){Wp[ka * OUT + n],   Wp[(ka + 1) * OUT + n]};   WMMA4(accP[t],  a, b);
      b = (v2f){Was[ka * OUT + n],  Was[(ka + 1) * OUT + n]};  WMMA4(accAS[t], a, b);
      b = (v2f){Wad[ka * OUT + n],  Wad[(ka + 1) * OUT + n]};  WMMA4(accAD[t], a, b);
      b = (v2f){Wapd[ka * OUT + n], Wapd[(ka + 1) * OUT + n]}; WMMA4(accAP[t], a, b);
    }
  }

  // C-layout -> LDS (row-major 16x32) with leaky, for the chained Wp2 GEMM
  __shared__ float lds[16 * OUT];
#pragma unroll
  for (int t = 0; t < 2; ++t)
#pragma unroll
    for (int v = 0; v < 8; ++v)
      lds[(v + 8 * hi) * OUT + lo + 16 * t] = leaky(accP[t][v]);
  __syncthreads();

  v8f accFC[2] = {};
  for (int k0 = 0; k0 < OUT; k0 += 4) {
    const int ka = k0 + 2 * hi;
    v2f a2 = (v2f){lds[lo * OUT + ka], lds[lo * OUT + ka + 1]};
#pragma unroll
    for (int t = 0; t < 2; ++t) {
      const int n = lo + 16 * t;
      v2f b = (v2f){Wp2[ka * OUT + n], Wp2[(ka + 1) * OUT + n]};
      WMMA4(accFC[t], a2, b);
    }
  }

#pragma unroll
  for (int t = 0; t < 2; ++t)
#pragma unroll
    for (int v = 0; v < 8; ++v) {
      const long row = base + v + 8 * hi;
      const int n = lo + 16 * t;
      fc[row * OUT + n]    = accFC[t][v];
      esrc[row * OUT + n]  = accAS[t][v];
      edst[row * OUT + n]  = accAD[t][v];
      dterm[row * OUT + n] = accAP[t][v] + bapd[n];
    }
}

// ---------------- kernel 2: edge logits + segment max (one wave = 16 edges) -
__global__ __launch_bounds__(32) void edge_logits_kernel(
    const float* __restrict__ fe, const int* __restrict__ src,
    const int* __restrict__ dst, const float* __restrict__ We,
    const float* __restrict__ esrc, const float* __restrict__ edst,
    float* __restrict__ ebuf, unsigned* __restrict__ mdst,
    unsigned* __restrict__ msrc) {
  const int lane = threadIdx.x, lo = lane & 15, hi = lane >> 4;
  const int ebase = blockIdx.x * 16;

  v8f acc[2] = {};
#pragma unroll
  for (int k0 = 0; k0 < F_E; k0 += 4) {
    const int ka = k0 + 2 * hi;
    v2f a = *(const v2f*)(fe + (long)(ebase + lo) * F_E + ka);
#pragma unroll
    for (int t = 0; t < 2; ++t) {
      const int n = lo + 16 * t;
      v2f b = (v2f){We[ka * OUT + n], We[(ka + 1) * OUT + n]};
      WMMA4(acc[t], a, b);
    }
  }

#pragma unroll
  for (int v = 0; v < 8; ++v) {
    const long eid = ebase + v + 8 * hi;
    const long s = src[eid], d = dst[eid];
#pragma unroll
    for (int t = 0; t < 2; ++t) {
      const int n = lo + 16 * t;
      float val = leaky(acc[t][v] + esrc[s * OUT + n] + edst[d * OUT + n]);
      ebuf[eid * OUT + n] = val;
      const unsigned fv = flipf(val);
      atomicMax(&mdst[d * OUT + n], fv);
      atomicMax(&msrc[s * OUT + n], fv);
    }
  }
}

// ---------------- kernel 3: segment softmax denominators --------------------
__global__ __launch_bounds__(256) void edge_sum_kernel(
    const float* __restrict__ ebuf, const int* __restrict__ src,
    const int* __restrict__ dst, const unsigned* __restrict__ mdst,
    const unsigned* __restrict__ msrc, float* __restrict__ sdst,
    float* __restrict__ ssrc) {
  const int idx = blockIdx.x * 256 + threadIdx.x;
  if (idx >= N_EDGES * OUT) return;
  const int eid = idx >> 5, n = idx & 31;
  const long s = src[eid], d = dst[eid];
  const float v = ebuf[idx];
  atomicAdd(&sdst[d * OUT + n], __expf(v - unflipf(mdst[d * OUT + n])));
  atomicAdd(&ssrc[s * OUT + n], __expf(v - unflipf(msrc[s * OUT + n])));
}

// ---------------- kernel 4: both-softmax attention + message scatter --------
__global__ __launch_bounds__(256) void edge_agg_kernel(
    const float* __restrict__ ebuf, const int* __restrict__ src,
    const int* __restrict__ dst, const unsigned* __restrict__ mdst,
    const unsigned* __restrict__ msrc, const float* __restrict__ sdst,
    const float* __restrict__ ssrc, const float* __restrict__ fc,
    float* __restrict__ msg) {
  const int idx = blockIdx.x * 256 + threadIdx.x;
  if (idx >= N_EDGES * OUT) return;
  const int eid = idx >> 5, n = idx & 31;
  const long s = src[eid], d = dst[eid];
  const float v = ebuf[idx];
  const float ad = fmaxf(__expf(v - unflipf(mdst[d * OUT + n])) / sdst[d * OUT + n], EPS);
  const float as = fmaxf(__expf(v - unflipf(msrc[s * OUT + n])) / ssrc[s * OUT + n], EPS);
  atomicAdd(&msg[d * OUT + n], fc[s * OUT + n] * sqrtf(ad * as));
}

// ---------------- kernel 5: per-head normalization (in place on msg) --------
__global__ __launch_bounds__(256) void norm_kernel(
    float* __restrict__ msg, const float* __restrict__ scale,
    const float* __restrict__ offset) {
  const int node = blockIdx.x * 256 + threadIdx.x;
  if (node >= N_NODES) return;
  float x[OUT];
#pragma unroll
  for (int i = 0; i < OUT; ++i) x[i] = msg[(long)node * OUT + i];
#pragma unroll
  for (int h = 0; h < 4; ++h) {
    float m = 0.f;
#pragma unroll
    for (int c = 0; c < 8; ++c) m += x[h * 8 + c];
    m *= 0.125f;
    float var = 0.f;
#pragma unroll
    for (int c = 0; c < 8; ++c) { float dd = x[h * 8 + c] - m; var += dd * dd; }
    const float r = rsqrtf(var * 0.125f + EPS);
#pragma unroll
    for (int c = 0; c < 8; ++c) {
      const int n = h * 8 + c;
      msg[(long)node * OUT + n] = (x[n] - m) * scale[n] * r + offset[n];
    }
  }
}

// ---------------- kernel 6: epilogue GEMMs (one wave = 16 rows) -------------
// out = leaky(msg @ Wagg + bagg + dterm) @ Wap + bap
__global__ __launch_bounds__(32) void final_kernel(
    const float* __restrict__ msg, const float* __restrict__ Wagg,
    const float* __restrict__ bagg, const float* __restrict__ dterm,
    const float* __restrict__ Wap, const float* __restrict__ bap,
    float* __restrict__ out) {
  const int lane = threadIdx.x, lo = lane & 15, hi = lane >> 4;
  const int base = blockIdx.x * 16;

  v8f acc1[2] = {};
  for (int k0 = 0; k0 < OUT; k0 += 4) {
    const int ka = k0 + 2 * hi;
    v2f a = *(const v2f*)(msg + (long)(base + lo) * OUT + ka);
#pragma unroll
    for (int t = 0; t < 2; ++t) {
      const int n = lo + 16 * t;
      v2f b = (v2f){Wagg[ka * OUT + n], Wagg[(ka + 1) * OUT + n]};
      WMMA4(acc1[t], a, b);
    }
  }

  __shared__ float lds[16 * OUT];
#pragma unroll
  for (int t = 0; t < 2; ++t)
#pragma unroll
    for (int v = 0; v < 8; ++v) {
      const long row = base + v + 8 * hi;
      const int n = lo + 16 * t;
      lds[(v + 8 * hi) * OUT + n] = leaky(acc1[t][v] + bagg[n] + dterm[row * OUT + n]);
    }
  __syncthreads();

  v8f acc2[2] = {};
  for (int k0 = 0; k0 < OUT; k0 += 4) {
    const int ka = k0 + 2 * hi;
    v2f a2 = (v2f){lds[lo * OUT + ka], lds[lo * OUT + ka + 1]};
#pragma unroll
    for (int t = 0; t < 2; ++t) {
      const int n = lo + 16 * t;
      v2f b = (v2f){Wap[ka * OUT + n], Wap[(ka + 1) * OUT + n]};
      WMMA4(acc2[t], a2, b);
    }
  }

#pragma unroll
  for (int t = 0; t < 2; ++t)
#pragma unroll
    for (int v = 0; v < 8; ++v)
      out[(long)(base + v + 8 * hi) * OUT + lo + 16 * t] = acc2[t][v] + bap[lo + 16 * t];
}

// ---------------- host launcher --------------------------------------------
extern "C" void kernel_launch(void* const* d_in, const int* in_sizes, int n_in,
                              void* d_out, int out_size, void* d_ws, size_t ws_size,
                              hipStream_t stream) {
  const float* feat  = (const float*)d_in[0];
  const float* fe    = (const float*)d_in[1];
  const int*   src   = (const int*)d_in[2];
  const int*   dst   = (const int*)d_in[3];
  const float* Wp    = (const float*)d_in[4];
  const float* Wp2   = (const float*)d_in[5];
  const float* Was   = (const float*)d_in[6];
  const float* Wad   = (const float*)d_in[7];
  const float* We    = (const float*)d_in[8];
  const float* scale = (const float*)d_in[9];
  const float* offs  = (const float*)d_in[10];
  const float* Wagg  = (const float*)d_in[11];
  const float* bagg  = (const float*)d_in[12];
  const float* Wapd  = (const float*)d_in[13];
  const float* bapd  = (const float*)d_in[14];
  const float* Wap   = (const float*)d_in[15];
  const float* bap   = (const float*)d_in[16];
  float* out = (float*)d_out;

  // workspace layout (floats): 9 node arrays of N*32 + edge buffer E*32
  const long NW = (long)N_NODES * OUT;
  float* fc    = (float*)d_ws;
  float* esrc  = fc + NW;
  float* edst  = esrc + NW;
  float* dterm = edst + NW;
  unsigned* mdst = (unsigned*)(dterm + NW);   // flipped-uint seg-max
  unsigned* msrc = mdst + NW;
  float* sdst  = (float*)(msrc + NW);
  float* ssrc  = sdst + NW;
  float* msg   = ssrc + NW;
  float* ebuf  = msg + NW;                    // E*32 floats

  // zero: mdst..msg (5 arrays). 0u is the flip-space -inf sentinel.
  const long nzero = 5 * NW;
  zero_kernel<<<(int)((nzero + 255) / 256), 256, 0, stream>>>((float*)mdst, nzero);

  node_gemm_kernel<<<N_NODES / 16, 32, 0, stream>>>(
      feat, Wp, Wp2, Was, Wad, Wapd, bapd, fc, esrc, edst, dterm);

  edge_logits_kernel<<<N_EDGES / 16, 32, 0, stream>>>(
      fe, src, dst, We, esrc, edst, ebuf, mdst, msrc);

  const int ne = N_EDGES * OUT;
  edge_sum_kernel<<<(ne + 255) / 256, 256, 0, stream>>>(
      ebuf, src, dst, mdst, msrc, sdst, ssrc);

  edge_agg_kernel<<<(ne + 255) / 256, 256, 0, stream>>>(
      ebuf, src, dst, mdst, msrc, sdst, ssrc, fc, msg);

  norm_kernel<<<(N_NODES + 255) / 256, 256, 0, stream>>>(msg, scale, offs);

  final_kernel<<<N_NODES / 16, 32, 0, stream>>>(
      msg, Wagg, bagg, dterm, Wap, bap, out);

  (void)in_sizes; (void)n_in; (void)out_size; (void)ws_size;
}